// WMSELoss_Ratio_wforces_73160472920742
// MI455X (gfx1250) — compile-verified
//
#include <hip/hip_runtime.h>

// WMSELoss_Ratio_wforces — MI455X (gfx1250, wave32)
// Memory-bound streaming reduction: ~448 MB reads => ~19us floor @ 23.3 TB/s.
// Pass 1: min(en) via order-independent atomicMin on monotonic uint key.
// Pass 2: fused single pass over forces/forces_pred (B128 loads, NT temporal
//         hint so the 432MB stream doesn't evict the L2-resident en) with
//         w[n] distributed over elements; energy term in same kernel.
//         Wave sums done on the matrix pipe (v_wmma_f32_16x16x4_f32).
// Pass 3: deterministic fixed-tree reduction of block partials -> scalar.

#define NCONF  2000000
#define KPC    27              // NATOMS*3 elements per config
#define GRID1  1024
#define GRID2  2048
#define BLK    256

typedef __attribute__((ext_vector_type(2))) float v2f;
typedef __attribute__((ext_vector_type(4))) float v4f;   // native clang vector: OK for NT builtin
typedef __attribute__((ext_vector_type(8))) float v8f;

__device__ __forceinline__ unsigned int f32_key(float f) {
    unsigned int u = __float_as_uint(f);
    return (u & 0x80000000u) ? ~u : (u | 0x80000000u);   // monotonic for min
}
__device__ __forceinline__ float key_f32(unsigned int k) {
    unsigned int u = (k & 0x80000000u) ? (k ^ 0x80000000u) : ~k;
    return __uint_as_float(u);
}

// Wave32 sum on the CDNA5 matrix pipe.
// A: lane puts acc in K-slot0 (lanes 0-15 -> K0, lanes 16-31 -> K2 per ISA
// 16x4 f32 A layout), zero in the other slot. B = ones(4x16).
// D[m,n] = acc[m] + acc[m+16]  (row-sum broadcast across columns).
// Sum of the 8 D VGPRs gives half-sums per 16-lane half; xor-16 finishes.
__device__ __forceinline__ float wave_sum_wmma(float v) {
    v2f a; a[0] = v;    a[1] = 0.0f;
    v2f b; b[0] = 1.0f; b[1] = 1.0f;
    v8f c = {};
    v8f d = __builtin_amdgcn_wmma_f32_16x16x4_f32(
        /*neg_a=*/false, a, /*neg_b=*/false, b,
        /*c_mod=*/(short)0, c, /*reuse_a=*/false, /*reuse_b=*/false);
    float s = ((d[0] + d[1]) + (d[2] + d[3])) + ((d[4] + d[5]) + (d[6] + d[7]));
    return s + __shfl_xor(s, 16, 32);
}

__global__ void kinit(unsigned int* minkey) {
    *minkey = 0xFFFFFFFFu;                // key of +max => identity for min
}

__global__ void kmin(const float* __restrict__ en, unsigned int* __restrict__ minkey) {
    const v4f* en4 = (const v4f*)en;
    const int n4 = NCONF / 4;
    float m = 3.0e38f;
    for (int i = blockIdx.x * blockDim.x + threadIdx.x; i < n4; i += gridDim.x * blockDim.x) {
        v4f e = en4[i];
        m = fminf(m, fminf(fminf(e[0], e[1]), fminf(e[2], e[3])));
    }
    #pragma unroll
    for (int off = 16; off; off >>= 1) m = fminf(m, __shfl_xor(m, off, 32));
    __shared__ float sm[BLK / 32];
    const int lane = threadIdx.x & 31, wid = threadIdx.x >> 5;
    if (lane == 0) sm[wid] = m;
    __syncthreads();
    if (threadIdx.x == 0) {
        float bm = sm[0];
        #pragma unroll
        for (int w = 1; w < BLK / 32; ++w) bm = fminf(bm, sm[w]);
        atomicMin(minkey, f32_key(bm));   // order-independent => deterministic
    }
}

__global__ void kmain(const float* __restrict__ en, const float* __restrict__ en_pred,
                      const float* __restrict__ F, const float* __restrict__ G,
                      const float* __restrict__ mean_p, const float* __restrict__ std_p,
                      const unsigned int* __restrict__ minkey,
                      float* __restrict__ partials) {
    const float mean = mean_p[0];
    const float stdv = std_p[0];
    const float enmin = fmaf(stdv, key_f32(*minkey), mean);   // min of de-normalized en
    const int tid = blockIdx.x * blockDim.x + threadIdx.x;
    const int stride = gridDim.x * blockDim.x;

    // ---- energy term: sum w * (_en - _en_pred)^2 ----
    float accE = 0.0f;
    {
        const v4f* en4 = (const v4f*)en;
        const v4f* ep4 = (const v4f*)en_pred;
        for (int i = tid; i < NCONF / 4; i += stride) {
            v4f e = en4[i];                                   // RT: keep en cached
            v4f p = __builtin_nontemporal_load(&ep4[i]);      // read-once stream
            #pragma unroll
            for (int j = 0; j < 4; ++j) {
                float de = fmaf(stdv, e[j], mean);
                float w  = 1.0f / (1.0f + (de - enmin));      // DWT = 1.0
                float dd = stdv * (e[j] - p[j]);
                accE = fmaf(w * dd, dd, accE);
            }
        }
    }

    // ---- force term: sum over all 54M elements of f*g*w[n] (w distributed) ----
    float accF = 0.0f;
    {
        const v4f* F4 = (const v4f*)F;
        const v4f* G4 = (const v4f*)G;
        const int ne4 = (NCONF * KPC) / 4;                    // 13.5M x B128
        for (int i = tid; i < ne4; i += stride) {
            const unsigned int e0 = 4u * (unsigned int)i;
            const unsigned int n0 = e0 / 27u;                 // magic-mul division
            const unsigned int k0 = e0 - 27u * n0;
            v4f f = __builtin_nontemporal_load(&F4[i]);       // NT: don't evict en
            v4f g = __builtin_nontemporal_load(&G4[i]);       // NT: don't evict en
            float de0 = fmaf(stdv, en[n0], mean);             // L2-resident re-reads
            float w0  = 1.0f / (1.0f + (de0 - enmin));
            unsigned int n1 = n0 + 1u; if (n1 >= NCONF) n1 = NCONF - 1u;
            float de1 = fmaf(stdv, en[n1], mean);
            float w1  = 1.0f / (1.0f + (de1 - enmin));
            // a B128 spans at most one config boundary (4 < 27)
            float wb = (k0 < 26u) ? w0 : w1;
            float wc = (k0 < 25u) ? w0 : w1;
            float wd = (k0 < 24u) ? w0 : w1;
            accF = fmaf(f[0] * g[0], w0, accF);
            accF = fmaf(f[1] * g[1], wb, accF);
            accF = fmaf(f[2] * g[2], wc, accF);
            accF = fmaf(f[3] * g[3], wd, accF);
        }
    }

    float total = fmaf(accF, (1.0f / 9.0f), accE);            // /NATOMS folded

    // Wave reduction on the matrix pipe (EXEC all-1s here: no divergence).
    float wsum = wave_sum_wmma(total);

    __shared__ float sm[BLK / 32];
    const int lane = threadIdx.x & 31, wid = threadIdx.x >> 5;
    if (lane == 0) sm[wid] = wsum;
    __syncthreads();
    if (threadIdx.x == 0) {
        float b = 0.0f;
        #pragma unroll
        for (int w = 0; w < BLK / 32; ++w) b += sm[w];
        partials[blockIdx.x] = b;
    }
}

__global__ void kfinal(const float* __restrict__ partials, float* __restrict__ out) {
    __shared__ double sm[BLK];
    double a = 0.0;
    for (int i = threadIdx.x; i < GRID2; i += BLK) a += (double)partials[i];
    sm[threadIdx.x] = a;
    __syncthreads();
    for (int s = BLK / 2; s > 0; s >>= 1) {                   // fixed tree: deterministic
        if (threadIdx.x < s) sm[threadIdx.x] += sm[threadIdx.x + s];
        __syncthreads();
    }
    if (threadIdx.x == 0) out[0] = (float)(sm[0] / (double)NCONF);
}

extern "C" void kernel_launch(void* const* d_in, const int* in_sizes, int n_in,
                              void* d_out, int out_size, void* d_ws, size_t ws_size,
                              hipStream_t stream) {
    (void)in_sizes; (void)n_in; (void)out_size; (void)ws_size;
    const float* en      = (const float*)d_in[0];
    const float* en_pred = (const float*)d_in[1];
    const float* F       = (const float*)d_in[2];   // forces      [N,9,3]
    const float* G       = (const float*)d_in[3];   // forces_pred [N,27]
    const float* mean_p  = (const float*)d_in[4];
    const float* std_p   = (const float*)d_in[5];

    unsigned int* minkey = (unsigned int*)d_ws;
    float* partials      = (float*)((char*)d_ws + 256);       // GRID2 floats

    kinit <<<1,     1,   0, stream>>>(minkey);
    kmin  <<<GRID1, BLK, 0, stream>>>(en, minkey);
    kmain <<<GRID2, BLK, 0, stream>>>(en, en_pred, F, G, mean_p, std_p, minkey, partials);
    kfinal<<<1,     BLK, 0, stream>>>(partials, (float*)d_out);
}